// ConvNat_54030688583934
// MI455X (gfx1250) — compile-verified
//
#include <hip/hip_runtime.h>
#include <hip/hip_bf16.h>

// ---------------------------------------------------------------------------
// ConvNat block for MI455X (gfx1250), wave32.
//   - Dense GEMMs (QKV / proj / final linear) via V_WMMA_F32_16X16X4_F32,
//     fully unrolled (compile-time N,K) -> 16 straight-line WMMAs per tile.
//   - Neighborhood attention (31x31 window, softmax) as wave-per-(pixel,head)
//     VALU code with LDS-staged probabilities and shuffle reductions.
// Shapes are compile-time constants from the reference: H=W=36, C=64,
// heads=4, dh=16, K=31.
// ---------------------------------------------------------------------------

typedef __attribute__((ext_vector_type(2))) float v2f;
typedef __attribute__((ext_vector_type(8))) float v8f;

#define NAT_H     36
#define NAT_W     36
#define NAT_C     64
#define NAT_HEADS 4
#define NAT_DH    16
#define NAT_K     31
#define NAT_K2    (NAT_K * NAT_K)     // 961
#define NAT_N     (NAT_H * NAT_W)     // 1296
#define RPB_W     (2 * NAT_K - 1)     // 61

// ---------------------------------------------------------------------------
// f32 WMMA GEMM:  Out[M,N] = A[M,K] @ Wt[N,K]^T + bias[N]
// One 16x16 output tile per wave32.  All dims compile-time; M=1296.
// A-fragment layout (ISA 7.12.2, 32-bit A 16x4):
//   lane L: m = L&15, a[r] = A[m][k0 + (L>>4)*2 + r]
// B-fragment (4x16, rows striped across lane halves, mirrored):
//   lane L: n = L&15, b[r] = B[k0 + (L>>4)*2 + r][n] = Wt[n][k0 + (L>>4)*2 + r]
// C/D: VGPR r holds (m = r + 8*(L>>4), n = L&15).
// ---------------------------------------------------------------------------
template <int M, int N, int K>
__global__ __launch_bounds__(256) void convnat_gemm_wmma_f32(
    const float* __restrict__ A, const float* __restrict__ Wt,
    const float* __restrict__ bias, float* __restrict__ Out) {
  constexpr int NTILES = N / 16;
  constexpr int TOTAL = (M / 16) * NTILES;

  const int wave = blockIdx.x * (blockDim.x >> 5) + (threadIdx.x >> 5);
  const int lane = threadIdx.x & 31;
  if (wave >= TOTAL) return;  // wave-uniform: EXEC stays all-ones for WMMA

  const int mt = wave / NTILES;          // constant divisor -> magic multiply
  const int nt = wave - mt * NTILES;
  const int m0 = mt << 4, n0 = nt << 4;
  const int lo = lane & 15, hi = lane >> 4;

  const float* __restrict__ arow = A + (m0 + lo) * K + hi * 2;
  const float* __restrict__ brow = Wt + (n0 + lo) * K + hi * 2;

  v8f c = {0.f, 0.f, 0.f, 0.f, 0.f, 0.f, 0.f, 0.f};
#pragma unroll
  for (int k0 = 0; k0 < K; k0 += 4) {
    v2f a, b;
    a[0] = arow[k0 + 0];
    a[1] = arow[k0 + 1];
    b[0] = brow[k0 + 0];
    b[1] = brow[k0 + 1];
    // (neg_a, A, neg_b, B, c_mod, C, reuse_a, reuse_b)
    c = __builtin_amdgcn_wmma_f32_16x16x4_f32(false, a, false, b, (short)0, c,
                                              false, false);
  }

  const float bv = bias[n0 + lo];
  float* __restrict__ orow = Out + (m0 + hi * 8) * N + (n0 + lo);
#pragma unroll
  for (int r = 0; r < 8; ++r) {
    orow[r * N] = c[r] + bv;
  }
}

// ---------------------------------------------------------------------------
// Neighborhood attention core.  qkv: (1296, 192) laid out [q(4x16)|k|v].
// One wave per (pixel, head).  8 waves / 256-thread block, grid = 648 exact.
// ---------------------------------------------------------------------------
__global__ __launch_bounds__(256) void convnat_nat_attn(
    const float* __restrict__ qkv, const float* __restrict__ rpb,
    float* __restrict__ out) {
  __shared__ float probs[8][NAT_K2];  // 30752 B

  const int wslot = threadIdx.x >> 5;
  const int lane = threadIdx.x & 31;
  const int wave = blockIdx.x * 8 + wslot;   // grid sized exactly: 5184 waves
  const int pix = wave >> 2;                 // / HEADS
  const int h = wave & 3;
  const int i = pix / NAT_W, j = pix - (pix / NAT_W) * NAT_W;

  int ri = i - NAT_K / 2;
  ri = ri < 0 ? 0 : (ri > NAT_H - NAT_K ? NAT_H - NAT_K : ri);
  int ci = j - NAT_K / 2;
  ci = ci < 0 ? 0 : (ci > NAT_W - NAT_K ? NAT_W - NAT_K : ci);

  // q (scaled by dh^-0.5 = 0.25), broadcast to every lane of the wave
  const float* __restrict__ qp = qkv + (size_t)pix * (3 * NAT_C) + h * NAT_DH;
  float q[NAT_DH];
#pragma unroll
  for (int d = 0; d < NAT_DH; ++d) q[d] = qp[d] * 0.25f;

  // rpb slice base: rrel = (ri + r) - i + 30, crel = (ci + c) - j + 30
  const float* __restrict__ rp =
      rpb + (size_t)h * RPB_W * RPB_W + (ri - i + NAT_K - 1) * RPB_W +
      (ci - j + NAT_K - 1);

  // ---- phase 1: logits + running max ----
  float lmax = -1e30f;
  for (int n = lane; n < NAT_K2; n += 32) {
    const int r = n / NAT_K;
    const int cc = n - r * NAT_K;
    const float* __restrict__ kp =
        qkv + (size_t)((ri + r) * NAT_W + (ci + cc)) * (3 * NAT_C) + NAT_C +
        h * NAT_DH;
    float dot = 0.f;
#pragma unroll
    for (int d = 0; d < NAT_DH; ++d) dot += q[d] * kp[d];
    dot += rp[r * RPB_W + cc];
    probs[wslot][n] = dot;
    lmax = fmaxf(lmax, dot);
  }
#pragma unroll
  for (int off = 16; off > 0; off >>= 1)
    lmax = fmaxf(lmax, __shfl_xor(lmax, off, 32));
  __syncthreads();

  // ---- phase 2: exp + sum ----
  float lsum = 0.f;
  for (int n = lane; n < NAT_K2; n += 32) {
    const float e = __expf(probs[wslot][n] - lmax);
    probs[wslot][n] = e;
    lsum += e;
  }
#pragma unroll
  for (int off = 16; off > 0; off >>= 1) lsum += __shfl_xor(lsum, off, 32);
  const float inv = 1.f / lsum;
  __syncthreads();

  // ---- phase 3: weighted sum of v ----
  // lane owns output dim d = lane&15; lane-halves split even/odd neighbors.
  const int d = lane & 15;
  const int half = lane >> 4;
  float acc = 0.f;
  for (int n = half; n < NAT_K2; n += 2) {
    const int r = n / NAT_K;
    const int cc = n - r * NAT_K;
    acc += probs[wslot][n] *
           qkv[(size_t)((ri + r) * NAT_W + (ci + cc)) * (3 * NAT_C) +
               2 * NAT_C + h * NAT_DH + d];
  }
  acc += __shfl_xor(acc, 16, 32);
  if (lane < 16) out[(size_t)pix * NAT_C + h * NAT_DH + d] = acc * inv;
}

// ---------------------------------------------------------------------------
// Depthwise 3x3 conv on the ORIGINAL input (NHWC) + add NAT output:
//   out[pix][c] = nat[pix][c] + dw_b[c] + sum_{3x3 valid} x * dw_w[c,0,di,dj]
// ---------------------------------------------------------------------------
__global__ __launch_bounds__(256) void convnat_dwconv_add(
    const float* __restrict__ x, const float* __restrict__ w,
    const float* __restrict__ b, const float* __restrict__ nat,
    float* __restrict__ out) {
  const int tid = blockIdx.x * blockDim.x + threadIdx.x;
  if (tid >= NAT_N * NAT_C) return;
  const int c = tid & (NAT_C - 1);
  const int pix = tid >> 6;
  const int i = pix / NAT_W, j = pix - (pix / NAT_W) * NAT_W;

  float s = b[c];
#pragma unroll
  for (int di = 0; di < 3; ++di) {
    const int ii = i + di - 1;
    if (ii < 0 || ii >= NAT_H) continue;
#pragma unroll
    for (int dj = 0; dj < 3; ++dj) {
      const int jj = j + dj - 1;
      if (jj < 0 || jj >= NAT_W) continue;
      s += x[(size_t)(ii * NAT_W + jj) * NAT_C + c] * w[c * 9 + di * 3 + dj];
    }
  }
  out[tid] = nat[tid] + s;
}

// ---------------------------------------------------------------------------
// Orchestration
// ---------------------------------------------------------------------------
extern "C" void kernel_launch(void* const* d_in, const int* in_sizes, int n_in,
                              void* d_out, int out_size, void* d_ws,
                              size_t ws_size, hipStream_t stream) {
  const float* x       = (const float*)d_in[0];
  const float* qkv_w1  = (const float*)d_in[3];
  const float* qkv_b1  = (const float*)d_in[4];
  const float* rpb1    = (const float*)d_in[5];
  const float* proj_w1 = (const float*)d_in[6];
  const float* proj_b1 = (const float*)d_in[7];
  const float* qkv_w2  = (const float*)d_in[8];
  const float* qkv_b2  = (const float*)d_in[9];
  const float* rpb2    = (const float*)d_in[10];
  const float* proj_w2 = (const float*)d_in[11];
  const float* proj_b2 = (const float*)d_in[12];
  const float* dw_w    = (const float*)d_in[13];
  const float* dw_b    = (const float*)d_in[14];
  const float* lin_w   = (const float*)d_in[15];
  const float* lin_b   = (const float*)d_in[16];
  float* out = (float*)d_out;

  // workspace carve-out (floats): ~2.3 MB total
  float* ws   = (float*)d_ws;
  float* qkvb = ws;                               // 1296*192 = 248832
  float* attn = qkvb + (size_t)NAT_N * 3 * NAT_C; // 1296*64
  float* nat1 = attn + (size_t)NAT_N * NAT_C;     // 1296*64
  float* pre  = nat1 + (size_t)NAT_N * NAT_C;     // 1296*64 (nat2 + conv)
  float* nat2 = pre + (size_t)NAT_N * NAT_C;      // 1296*64

  const int THREADS = 256;
  const int WPB = THREADS / 32;

  // GEMM tile counts
  const int qkv_tiles = (NAT_N / 16) * ((3 * NAT_C) / 16);  // 81*12 = 972
  const int c64_tiles = (NAT_N / 16) * (NAT_C / 16);        // 81*4  = 324
  const int qkv_blocks = (qkv_tiles + WPB - 1) / WPB;       // 122
  const int c64_blocks = (c64_tiles + WPB - 1) / WPB;       // 41
  const int attn_blocks = (NAT_N * NAT_HEADS) / WPB;        // 648 exact
  const int ew_blocks = (NAT_N * NAT_C + THREADS - 1) / THREADS;

  // ---- layer 1 ----
  convnat_gemm_wmma_f32<NAT_N, 3 * NAT_C, NAT_C>
      <<<qkv_blocks, THREADS, 0, stream>>>(x, qkv_w1, qkv_b1, qkvb);
  convnat_nat_attn<<<attn_blocks, THREADS, 0, stream>>>(qkvb, rpb1, attn);
  convnat_gemm_wmma_f32<NAT_N, NAT_C, NAT_C>
      <<<c64_blocks, THREADS, 0, stream>>>(attn, proj_w1, proj_b1, nat1);

  // ---- layer 2 ----
  convnat_gemm_wmma_f32<NAT_N, 3 * NAT_C, NAT_C>
      <<<qkv_blocks, THREADS, 0, stream>>>(nat1, qkv_w2, qkv_b2, qkvb);
  convnat_nat_attn<<<attn_blocks, THREADS, 0, stream>>>(qkvb, rpb2, attn);
  convnat_gemm_wmma_f32<NAT_N, NAT_C, NAT_C>
      <<<c64_blocks, THREADS, 0, stream>>>(attn, proj_w2, proj_b2, nat2);

  // ---- depthwise conv on original x + residual-style add ----
  convnat_dwconv_add<<<ew_blocks, THREADS, 0, stream>>>(x, dw_w, dw_b, nat2,
                                                        pre);

  // ---- final linear ----
  convnat_gemm_wmma_f32<NAT_N, NAT_C, NAT_C>
      <<<c64_blocks, THREADS, 0, stream>>>(pre, lin_w, lin_b, out);
}